// RadialFunction_77120432767612
// MI455X (gfx1250) — compile-verified
//
#include <hip/hip_runtime.h>
#include <stdint.h>
#include <stddef.h>

// ---------------------------------------------------------------------------
// RadialFunction: out[n, r] = cutoff(dr_n) * EMBED_NORM *
//                 sum_b embeddings[Zj_n, Zi_n, r, b] * RAD_NORM*exp(-B*(s_b-dr_n)^2)
// Memory-bound gather kernel: ~32 HBM B/edge + 140 L2 B/edge, ~90 flops/edge.
// CDNA5 paths used: async global->LDS double-buffered staging (ASYNCcnt),
// wide b128 gathers, non-temporal stores.
// ---------------------------------------------------------------------------

#define TILE     256
#define MAX_GRID 4096

constexpr int   kNB    = 7;                      // N_BASIS
constexpr int   kNR    = 5;                      // N_RADIAL
constexpr int   kNS    = 119;                    // N_SPECIES
constexpr float kRMax  = 6.0f;
constexpr float kRMin  = 0.5f;
constexpr float kBetta = 49.0f / 36.0f;          // N_BASIS^2 / R_MAX^2
constexpr float kCosK  = 0.52359877559829887f;   // pi / R_MAX
// RAD_NORM * EMBED_NORM = (2*BETTA/pi)^0.25 / sqrt(7)
constexpr float kNorm  = 0.36466527f;

// ------------------------- CDNA5 async helpers -----------------------------
#if defined(__HIP_DEVICE_COMPILE__)

typedef __attribute__((address_space(1))) int  gint_t;
typedef __attribute__((address_space(3))) int  lint_t;

__device__ __forceinline__ void async_load_b32_to_lds(const void* gsrc, void* ldst) {
#if __has_builtin(__builtin_amdgcn_global_load_async_to_lds_b32)
  __builtin_amdgcn_global_load_async_to_lds_b32(
      (gint_t*)(void*)gsrc,
      (lint_t*)ldst,
      0, 0);
#else
  unsigned lds_off = (unsigned)(size_t)(__attribute__((address_space(3))) void*)ldst;
  asm volatile("global_load_async_to_lds_b32 %0, %1, off"
               :: "v"(lds_off), "v"((unsigned long long)(uintptr_t)gsrc)
               : "memory");
#endif
}

template <int N>
__device__ __forceinline__ void wait_asynccnt_le() {
#if __has_builtin(__builtin_amdgcn_s_wait_asynccnt)
  __builtin_amdgcn_s_wait_asynccnt(N);
#else
  if constexpr (N == 0) {
    asm volatile("s_wait_asynccnt 0x0" ::: "memory");
  } else {
    asm volatile("s_wait_asynccnt 0x3" ::: "memory");
  }
#endif
  asm volatile("" ::: "memory");  // compiler ordering fence for the LDS reads
}

#else  // host pass: bodies never executed, keep them empty

__device__ __forceinline__ void async_load_b32_to_lds(const void*, void*) {}
template <int N> __device__ __forceinline__ void wait_asynccnt_le() {}

#endif

// ------------------------------- kernel ------------------------------------

__global__ __launch_bounds__(TILE)
void radial_function_kernel(const float* __restrict__ dr,
                            const int*   __restrict__ Zi,
                            const int*   __restrict__ Zj,
                            const float* __restrict__ emb,
                            float*       __restrict__ out,
                            int n)
{
  // Double-buffered staging of the streaming operands. Each lane owns the
  // slot it writes AND reads (slot == threadIdx.x), so waves are fully
  // decoupled: only the per-wave ASYNCcnt wait is required, no barriers.
  __shared__ float s_dr[2][TILE];
  __shared__ int   s_zi[2][TILE];
  __shared__ int   s_zj[2][TILE];

  const int tid   = threadIdx.x;
  const int tiles = (n + TILE - 1) / TILE;

  auto issue = [&](int t, int b) {
    int i = t * TILE + tid;
    if (i >= n) i = n - 1;                 // tail: clamp to a valid address
    async_load_b32_to_lds(dr + i, &s_dr[b][tid]);
    async_load_b32_to_lds(Zi + i, &s_zi[b][tid]);
    async_load_b32_to_lds(Zj + i, &s_zj[b][tid]);
  };

  int t = blockIdx.x;
  if (t >= tiles) return;

  int buf = 0;
  issue(t, 0);                             // prologue: prefetch first tile

  for (; t < tiles; t += gridDim.x) {
    const int tn = t + (int)gridDim.x;
    if (tn < tiles) {
      issue(tn, buf ^ 1);                  // kick next tile's async loads
      wait_asynccnt_le<3>();               // oldest 3 (current tile) complete
    } else {
      wait_asynccnt_le<0>();
    }

    const int idx = t * TILE + tid;
    if (idx < n) {
      const float d  = s_dr[buf][tid];
      const int   zi = s_zi[buf][tid];
      const int   zj = s_zj[buf][tid];

      // 140 B contiguous coefficient row; wide loads maximize gather MLP.
      const float* crow = emb + ((size_t)zj * kNS + (size_t)zi) * (kNR * kNB);
      float c[kNR * kNB];
      __builtin_memcpy(c, crow, sizeof(c));

      // Gaussian basis (v_exp_f32), normalization folded into kNorm.
      float e[kNB];
#pragma unroll
      for (int b = 0; b < kNB; ++b) {
        const float sh = kRMin + ((kRMax - kRMin) / (float)kNB) * (float)b;
        const float x  = sh - d;
        e[b] = __expf(-kBetta * x * x);
      }

      // cosine cutoff (v_cos_f32) * RAD_NORM * EMBED_NORM
      const float dc    = fminf(d, kRMax);
      const float scale = (0.5f * (__cosf(kCosK * dc) + 1.0f)) * kNorm;

      float* op = out + (size_t)idx * kNR;
#pragma unroll
      for (int r = 0; r < kNR; ++r) {
        float acc = 0.0f;
#pragma unroll
        for (int b = 0; b < kNB; ++b)
          acc = __builtin_fmaf(c[r * kNB + b], e[b], acc);
        // write-once 100 MB output: NT store keeps the 2 MB table hot in L2
        __builtin_nontemporal_store(acc * scale, op + r);
      }
    }
    buf ^= 1;
  }
}

// ------------------------------ launcher -----------------------------------

extern "C" void kernel_launch(void* const* d_in, const int* in_sizes, int n_in,
                              void* d_out, int out_size, void* d_ws, size_t ws_size,
                              hipStream_t stream) {
  const float* dr  = (const float*)d_in[0];
  const int*   Zi  = (const int*)  d_in[1];
  const int*   Zj  = (const int*)  d_in[2];
  const float* emb = (const float*)d_in[3];
  float*       out = (float*)d_out;

  const int n     = in_sizes[0];
  const int tiles = (n + TILE - 1) / TILE;
  const int grid  = tiles < MAX_GRID ? tiles : MAX_GRID;

  radial_function_kernel<<<grid, TILE, 0, stream>>>(dr, Zi, Zj, emb, out, n);
}